// LogBessel_7404523618944
// MI455X (gfx1250) — compile-verified
//
#include <hip/hip_runtime.h>
#include <math.h>

// log I_nu(kappa), nu = 31 (DIM/2 - 1), matching the JAX reference:
//   kappa_safe = max(kappa, 1e-10)
//   log_iv     = nu*log(kappa/2) - lgamma(nu+1) + log(sum_k r_k)
//   out        = log(exp(log_iv) + 1e-10)
// with r_0 = 1, r_k = r_{k-1} * q / (k*(k+31)), q = (kappa/2)^2.
// 49 terms (k=0..48) are exact to fp32 epsilon for kappa < 50
// (r_36/sum ~ 3e-9 at kappa=50); the reference's extra terms are sub-ulp.
//
// All log-domain math is done in base 2 so the hardware v_log_f32 /
// v_exp_f32 are used raw (arguments are >= 1e-10, far above the denormal
// threshold, so the libm denorm-rescale path is provably dead code).

__device__ __forceinline__ float fast_log2(float x) {
#if defined(__AMDGCN__) && __has_builtin(__builtin_amdgcn_logf)
    return __builtin_amdgcn_logf(x);      // v_log_f32
#else
    return log2f(x);                       // host-pass parse only
#endif
}

__device__ __forceinline__ float fast_exp2(float x) {
#if defined(__AMDGCN__) && __has_builtin(__builtin_amdgcn_exp2f)
    return __builtin_amdgcn_exp2f(x);     // v_exp_f32
#else
    return exp2f(x);                       // host-pass parse only
#endif
}

__device__ __forceinline__ float log_iv_nu31(float kappa) {
    const float EPSV      = 1e-10f;
    const float NU        = 31.0f;
    // log2(31!) = lgamma(32)/ln(2), folded at compile time
    const float LG32_LOG2 = (float)(78.09222355331531 / 0.6931471805599453);
    const float LN2       = 0.6931471805599453f;

    float x   = fmaxf(kappa, EPSV);
    float xh  = 0.5f * x;
    float l2x = fast_log2(xh);
    float q   = xh * xh;
    float q2  = q * q;

    // Two independent recurrence chains (odd/even k) for ILP.
    // r1 = q / (1*32); r2 = r1 * q / (2*33)
    float to = q  * (1.0f / 32.0f);
    float te = q2 * (1.0f / (32.0f * 66.0f));
    float so = to;
    float se = te;

#pragma unroll
    for (int j = 1; j <= 23; ++j) {
        // r_{2j+1} = r_{2j-1} * q^2 / ((2j)(2j+31)(2j+1)(2j+32))
        const float Ko = (float)(1.0 / ((double)(2 * j) * (double)(2 * j + 31)
                                      * (double)(2 * j + 1) * (double)(2 * j + 32)));
        // r_{2j+2} = r_{2j} * q^2 / ((2j+1)(2j+32)(2j+2)(2j+33))
        const float Ke = (float)(1.0 / ((double)(2 * j + 1) * (double)(2 * j + 32)
                                      * (double)(2 * j + 2) * (double)(2 * j + 33)));
        to *= q2 * Ko;
        te *= q2 * Ke;
        so += to;
        se += te;
    }

    float sum   = 1.0f + so + se;                       // Sigma r_k, k = 0..48
    float l2sum = fast_log2(sum);
    float l2iv  = fmaf(NU, l2x, l2sum - LG32_LOG2);     // log2 I_nu(x)
    float iv    = fast_exp2(l2iv);                      // I_nu(x) (0 on underflow)
    return LN2 * fast_log2(iv + EPSV);                  // ln(iv + eps)
}

__global__ __launch_bounds__(256) void logbessel_vec4(const float4* __restrict__ in,
                                                      float4* __restrict__ out,
                                                      int n4) {
    int i = blockIdx.x * blockDim.x + threadIdx.x;
    if (i >= n4) return;
    float4 k = in[i];       // global_load_b128, coalesced stream
    float4 r;
    r.x = log_iv_nu31(k.x); // 4 independent series + even/odd chain split
    r.y = log_iv_nu31(k.y); // -> 8-way ILP, lets the compiler keep packing
    r.z = log_iv_nu31(k.z); //    into v_pk_mul_f32 / dual-issue pairs
    r.w = log_iv_nu31(k.w);
    out[i] = r;             // global_store_b128
}

__global__ __launch_bounds__(64) void logbessel_tail(const float* __restrict__ in,
                                                     float* __restrict__ out,
                                                     int start, int n) {
    int i = start + blockIdx.x * blockDim.x + threadIdx.x;
    if (i < n) out[i] = log_iv_nu31(in[i]);
}

extern "C" void kernel_launch(void* const* d_in, const int* in_sizes, int n_in,
                              void* d_out, int out_size, void* d_ws, size_t ws_size,
                              hipStream_t stream) {
    (void)n_in; (void)out_size; (void)d_ws; (void)ws_size;
    const float* kappa = (const float*)d_in[0];
    float*       out   = (float*)d_out;
    int n  = in_sizes[0];          // 4096*4096 = 16,777,216
    int n4 = n >> 2;               // float4 elements

    if (n4 > 0) {
        int blocks = (n4 + 255) / 256;   // 16384 blocks of 8 wave32s
        logbessel_vec4<<<blocks, 256, 0, stream>>>(
            (const float4*)kappa, (float4*)out, n4);
    }
    int done = n4 << 2;
    if (done < n) {
        logbessel_tail<<<1, 64, 0, stream>>>(kappa, out, done, n);
    }
}